// PointTransformerCls_41051297415637
// MI455X (gfx1250) — compile-verified
//
#include <hip/hip_runtime.h>
#include <hip/hip_bf16.h>

// ---------------------------------------------------------------------------
// Point Transformer classification forward for gfx1250 (CDNA5, wave32).
// All dense linear algebra runs through V_WMMA_F32_16X16X4_F32 (fp32 WMMA).
// ---------------------------------------------------------------------------

typedef float v2f __attribute__((ext_vector_type(2)));
typedef float v8f __attribute__((ext_vector_type(8)));

#define BB 32
#define NPTS 2048
#define M1 512
#define M2 256

__device__ __forceinline__ float bn_apply(float x, float g, float b, float m, float v) {
    return (x - m) * g * rsqrtf(v + 1e-5f) + b;
}

// ---------------------------------------------------------------------------
// Input MLP: per-point 3 -> 64 -> 64, BN+ReLU each, output (B, N, 64) point-major
// ---------------------------------------------------------------------------
__global__ void k_inmlp(const float* __restrict__ x,
                        const float* __restrict__ w1, const float* g1, const float* b1,
                        const float* m1, const float* v1,
                        const float* __restrict__ w2, const float* g2, const float* b2,
                        const float* m2, const float* v2,
                        float* __restrict__ out, int total) {
    int id = blockIdx.x * blockDim.x + threadIdx.x;
    if (id >= total) return;
    float p0 = x[id * 3 + 0], p1 = x[id * 3 + 1], p2 = x[id * 3 + 2];
    float v[64];
    for (int o = 0; o < 64; ++o) {
        float t = w1[o * 3 + 0] * p0 + w1[o * 3 + 1] * p1 + w1[o * 3 + 2] * p2;
        t = bn_apply(t, g1[o], b1[o], m1[o], v1[o]);
        v[o] = fmaxf(t, 0.f);
    }
    for (int o = 0; o < 64; ++o) {
        float t = 0.f;
        for (int c = 0; c < 64; ++c) t += w2[o * 64 + c] * v[c];
        t = bn_apply(t, g2[o], b2[o], m2[o], v2[o]);
        out[(size_t)id * 64 + o] = fmaxf(t, 0.f);
    }
}

// ---------------------------------------------------------------------------
// Farthest point sampling. One 256-thread block per batch. Emits far BEFORE
// update (scan semantics: idx[0] == 0). Argmax ties -> lowest index.
// ---------------------------------------------------------------------------
__global__ __launch_bounds__(256) void k_fps(const float* __restrict__ xyz, int Npts,
                                             int npoint, int* __restrict__ idx_out) {
    const int b = blockIdx.x;
    xyz += (size_t)b * Npts * 3;
    idx_out += (size_t)b * npoint;
    const int tid = threadIdx.x;
    float dloc[8];
#pragma unroll
    for (int j = 0; j < 8; ++j) dloc[j] = 1e10f;
    __shared__ float sval[256];
    __shared__ int sidx[256];
    __shared__ int sfar;
    if (tid == 0) sfar = 0;
    const int cnt = (Npts + 255) >> 8;
    for (int step = 0; step < npoint; ++step) {
        __syncthreads();
        int far = sfar;
        if (tid == 0) idx_out[step] = far;
        float cx = xyz[far * 3 + 0], cy = xyz[far * 3 + 1], cz = xyz[far * 3 + 2];
        float best = -1.f;
        int bi = 0;
        for (int j = 0; j < cnt; ++j) {
            int i = tid + (j << 8);
            if (i < Npts) {
                float dx = xyz[i * 3 + 0] - cx;
                float dy = xyz[i * 3 + 1] - cy;
                float dz = xyz[i * 3 + 2] - cz;
                float d = dx * dx + dy * dy + dz * dz;
                float nd = fminf(dloc[j], d);
                dloc[j] = nd;
                if (nd > best) { best = nd; bi = i; }
            }
        }
        sval[tid] = best;
        sidx[tid] = bi;
        __syncthreads();
        for (int s = 128; s > 0; s >>= 1) {
            if (tid < s) {
                float ov = sval[tid + s];
                int oi = sidx[tid + s];
                if (ov > sval[tid] || (ov == sval[tid] && oi < sidx[tid])) {
                    sval[tid] = ov;
                    sidx[tid] = oi;
                }
            }
            __syncthreads();
        }
        if (tid == 0) sfar = sidx[0];
    }
}

// ---------------------------------------------------------------------------
// Gather 3D coords at sampled indices
// ---------------------------------------------------------------------------
__global__ void k_gather3(const float* __restrict__ src, const int* __restrict__ idx,
                          int Mq, int Np, float* __restrict__ dst, int total) {
    int id = blockIdx.x * blockDim.x + threadIdx.x;
    if (id >= total) return;
    int b = id / Mq;
    int pi = idx[id];
    const float* s = src + ((size_t)b * Np + pi) * 3;
    dst[id * 3 + 0] = s[0];
    dst[id * 3 + 1] = s[1];
    dst[id * 3 + 2] = s[2];
}

// ---------------------------------------------------------------------------
// kNN (K=32) matching lax.top_k(-d): ascending distance, lower index on tie.
// One thread per query, insertion sort on (d, idx).
// ---------------------------------------------------------------------------
__global__ void k_knn(const float* __restrict__ qxyz, const float* __restrict__ pxyz,
                      int Mq, int Np, int* __restrict__ out, int total) {
    int id = blockIdx.x * blockDim.x + threadIdx.x;
    if (id >= total) return;
    int b = id / Mq;
    const float* qp = qxyz + (size_t)id * 3;
    const float* pp = pxyz + (size_t)b * Np * 3;
    float qx = qp[0], qy = qp[1], qz = qp[2];
    float kd[32];
    int ki[32];
#pragma unroll
    for (int t = 0; t < 32; ++t) { kd[t] = 3.4e38f; ki[t] = 0x7fffffff; }
    for (int i = 0; i < Np; ++i) {
        float dx = pp[i * 3 + 0] - qx;
        float dy = pp[i * 3 + 1] - qy;
        float dz = pp[i * 3 + 2] - qz;
        float d = dx * dx + dy * dy + dz * dz;
        if (d < kd[31] || (d == kd[31] && i < ki[31])) {
            int pos = 31;
            while (pos > 0 && (d < kd[pos - 1] || (d == kd[pos - 1] && i < ki[pos - 1]))) {
                kd[pos] = kd[pos - 1];
                ki[pos] = ki[pos - 1];
                --pos;
            }
            kd[pos] = d;
            ki[pos] = i;
        }
    }
    int* op = out + (size_t)id * 32;
    for (int t = 0; t < 32; ++t) op[t] = ki[t];
}

// ---------------------------------------------------------------------------
// Fused local_op: gather group features [grouped-center ; center] (C=2*Ch rows,
// 32 cols) into LDS, then two WMMA GEMMs (W1, W2; each C x C) with BN+ReLU,
// then max over the 32 samples. One 128-thread (4-wave) block per group.
// Each wave owns a 16-row strip and computes BOTH 16-col tiles, sharing the
// A (weight) fragment across two WMMAs per k-step.
// ---------------------------------------------------------------------------
__global__ __launch_bounds__(128) void k_localop(
    const float* __restrict__ pts, int Nsrc, int Ch, int Mq,
    const int* __restrict__ fps, const int* __restrict__ knn,
    const float* __restrict__ W1, const float* g1, const float* b1,
    const float* m1, const float* v1,
    const float* __restrict__ W2, const float* g2, const float* b2,
    const float* m2, const float* v2,
    float* __restrict__ out) {
    __shared__ float Xs[256 * 32];
    __shared__ float Hs[256 * 32];
    const int C = 2 * Ch;
    const int g = blockIdx.x;
    const int b = g / Mq;
    const int tid = threadIdx.x;
    const int wid = tid >> 5;
    const int lane = tid & 31;
    const int lr = lane & 15;
    const int khi = (lane & 16) ? 2 : 0;
    const int rhi = (lane & 16) ? 8 : 0;
    const int* kn = knn + (size_t)g * 32;
    const int ci = fps[g];
    const float* ctrf = pts + ((size_t)b * Nsrc + ci) * Ch;

    for (int e = tid; e < C * 32; e += 128) {
        int kk = e >> 5, s = e & 31;
        float val;
        if (kk < Ch) {
            int pi = kn[s];
            val = pts[((size_t)b * Nsrc + pi) * Ch + kk] - ctrf[kk];
        } else {
            val = ctrf[kk - Ch];
        }
        Xs[kk * 32 + s] = val;
    }
    __syncthreads();

    const int RT = C >> 4;  // 16-row strips
    for (int pass = 0; pass < 2; ++pass) {
        const float* W = pass ? W2 : W1;
        const float* gg = pass ? g2 : g1;
        const float* bb = pass ? b2 : b1;
        const float* mm = pass ? m2 : m1;
        const float* vv = pass ? v2 : v1;
        const float* src = pass ? Hs : Xs;
        float* dst = pass ? Xs : Hs;
        for (int rt = wid; rt < RT; rt += 4) {
            const int mm0 = rt * 16;
            v8f acc0 = {};
            v8f acc1 = {};
#pragma unroll 4
            for (int k = 0; k < C; k += 4) {
                int ks = k + khi;
                v2f a, bv0, bv1;
                a.x = W[(mm0 + lr) * C + ks];
                a.y = W[(mm0 + lr) * C + ks + 1];
                bv0.x = src[ks * 32 + lr];
                bv0.y = src[(ks + 1) * 32 + lr];
                bv1.x = src[ks * 32 + 16 + lr];
                bv1.y = src[(ks + 1) * 32 + 16 + lr];
                acc0 = __builtin_amdgcn_wmma_f32_16x16x4_f32(false, a, false, bv0,
                                                             (short)0, acc0, false, false);
                acc1 = __builtin_amdgcn_wmma_f32_16x16x4_f32(false, a, false, bv1,
                                                             (short)0, acc1, false, false);
            }
#pragma unroll
            for (int r = 0; r < 8; ++r) {
                int orow = mm0 + r + rhi;
                float sg = gg[orow], sb = bb[orow], sm = mm[orow], sv = vv[orow];
                float v0 = fmaxf(bn_apply(acc0[r], sg, sb, sm, sv), 0.f);
                float v1 = fmaxf(bn_apply(acc1[r], sg, sb, sm, sv), 0.f);
                dst[orow * 32 + lr] = v0;
                dst[orow * 32 + 16 + lr] = v1;
            }
        }
        __syncthreads();
    }

    for (int c = tid; c < C; c += 128) {
        float mx = -3.4e38f;
        for (int s = 0; s < 32; ++s) mx = fmaxf(mx, Xs[c * 32 + s]);
        out[(size_t)g * C + c] = mx;
    }
}

// ---------------------------------------------------------------------------
// Generic batched WMMA-f32 GEMM: D = act(bn(op(A) @ op(B) + bias))
// One wave per 32x16 output tile (two accumulators sharing the B fragment),
// K stepped by 4, unrolled x4. Transpose flags are template params so each
// instantiation has a clean, branch-free WMMA inner loop. Address clamping
// keeps EXEC full through the WMMAs; stores are guarded.
// ---------------------------------------------------------------------------
template <int TA, int TB>
__global__ __launch_bounds__(32) void k_gemm(
    const float* __restrict__ A, const float* __restrict__ Bm, float* __restrict__ D,
    int M, int N, int K, int lda, int ldb, int ldd,
    long sA, long sB, long sD,
    const float* __restrict__ bias,
    const float* __restrict__ bg, const float* __restrict__ bb,
    const float* __restrict__ bm, const float* __restrict__ bv,
    int act) {
    const int lane = threadIdx.x & 31;
    const int lr = lane & 15;
    const int z = blockIdx.z;
    A += sA * z;
    Bm += sB * z;
    D += sD * z;
    const int m0 = blockIdx.y * 32, n0 = blockIdx.x * 16;
    const int row0 = m0 + lr;
    const int row1 = m0 + 16 + lr;
    const int rowc0 = row0 < M ? row0 : M - 1;
    const int rowc1 = row1 < M ? row1 : M - 1;
    const int col = n0 + lr;
    const int colc = col < N ? col : N - 1;
    const int khi = (lane & 16) ? 2 : 0;
    v8f acc0 = {};
    v8f acc1 = {};
#pragma unroll 4
    for (int k = 0; k < K; k += 4) {
        int ks = k + khi;
        v2f a0, a1, b;
        if (TA) {
            a0.x = A[(size_t)ks * lda + rowc0];
            a0.y = A[(size_t)(ks + 1) * lda + rowc0];
            a1.x = A[(size_t)ks * lda + rowc1];
            a1.y = A[(size_t)(ks + 1) * lda + rowc1];
        } else {
            a0.x = A[(size_t)rowc0 * lda + ks];
            a0.y = A[(size_t)rowc0 * lda + ks + 1];
            a1.x = A[(size_t)rowc1 * lda + ks];
            a1.y = A[(size_t)rowc1 * lda + ks + 1];
        }
        if (TB) {
            b.x = Bm[(size_t)colc * ldb + ks];
            b.y = Bm[(size_t)colc * ldb + ks + 1];
        } else {
            b.x = Bm[(size_t)ks * ldb + colc];
            b.y = Bm[(size_t)(ks + 1) * ldb + colc];
        }
        acc0 = __builtin_amdgcn_wmma_f32_16x16x4_f32(false, a0, false, b,
                                                     (short)0, acc0, false, false);
        acc1 = __builtin_amdgcn_wmma_f32_16x16x4_f32(false, a1, false, b,
                                                     (short)0, acc1, false, false);
    }
    const int ocol = n0 + lr;
#pragma unroll
    for (int half = 0; half < 2; ++half) {
        const v8f& acc = half ? acc1 : acc0;
        const int base = m0 + half * 16 + ((lane & 16) ? 8 : 0);
#pragma unroll
        for (int r = 0; r < 8; ++r) {
            int orow = base + r;
            if (orow < M && ocol < N) {
                float val = acc[r];
                if (bias) val += bias[orow];
                if (bg) val = bn_apply(val, bg[orow], bb[orow], bm[orow], bv[orow]);
                if (act == 1) val = fmaxf(val, 0.f);
                else if (act == 2) val = val > 0.f ? val : 0.2f * val;
                D[(size_t)orow * ldd + ocol] = val;
            }
        }
    }
}

// ---------------------------------------------------------------------------
// Positional bias: per (b,n,m): relu((xyz[n]-xyz[m]) @ pw1^T + pb1) @ pw2 + pb2
// ---------------------------------------------------------------------------
__global__ void k_pos(const float* __restrict__ xyz, const float* __restrict__ pw1,
                      const float* __restrict__ pb1, const float* __restrict__ pw2,
                      const float* __restrict__ pb2, float* __restrict__ pos,
                      int Np, int total) {
    int id = blockIdx.x * blockDim.x + threadIdx.x;
    if (id >= total) return;
    int m = id % Np;
    int t = id / Np;
    int n = t % Np;
    int b = t / Np;
    const float* xb = xyz + (size_t)b * Np * 3;
    float rx = xb[n * 3 + 0] - xb[m * 3 + 0];
    float ry = xb[n * 3 + 1] - xb[m * 3 + 1];
    float rz = xb[n * 3 + 2] - xb[m * 3 + 2];
    float acc = pb2[0];
#pragma unroll
    for (int h = 0; h < 16; ++h) {
        float hv = rx * pw1[h * 3 + 0] + ry * pw1[h * 3 + 1] + rz * pw1[h * 3 + 2] + pb1[h];
        acc += fmaxf(hv, 0.f) * pw2[h];
    }
    pos[id] = acc;
}

// ---------------------------------------------------------------------------
// Row softmax of (energy + pos), in place over energy. Block per row (len 256).
// ---------------------------------------------------------------------------
__global__ __launch_bounds__(256) void k_softmax(float* __restrict__ e,
                                                 const float* __restrict__ p, int L) {
    const size_t row = blockIdx.x;
    float* er = e + row * L;
    const float* pr = p + row * L;
    const int t = threadIdx.x;
    __shared__ float red[256];
    float v = er[t] + pr[t];
    red[t] = v;
    __syncthreads();
    for (int s = 128; s > 0; s >>= 1) {
        if (t < s) red[t] = fmaxf(red[t], red[t + s]);
        __syncthreads();
    }
    float mx = red[0];
    __syncthreads();
    float ex = expf(v - mx);
    red[t] = ex;
    __syncthreads();
    for (int s = 128; s > 0; s >>= 1) {
        if (t < s) red[t] += red[t + s];
        __syncthreads();
    }
    er[t] = ex / red[0];
}

// ---------------------------------------------------------------------------
// Column renormalization: att /= (1e-9 + sum over axis n). Block per (b, m).
// ---------------------------------------------------------------------------
__global__ __launch_bounds__(256) void k_colnorm(float* __restrict__ att, int Np) {
    const int m = blockIdx.x % Np;
    const int b = blockIdx.x / Np;
    float* base = att + (size_t)b * Np * Np;
    const int t = threadIdx.x;
    __shared__ float red[256];
    float v = base[(size_t)t * Np + m];
    red[t] = v;
    __syncthreads();
    for (int s = 128; s > 0; s >>= 1) {
        if (t < s) red[t] += red[t + s];
        __syncthreads();
    }
    base[(size_t)t * Np + m] = v / (1e-9f + red[0]);
}

__global__ void k_sub(const float* __restrict__ a, const float* __restrict__ b,
                      float* __restrict__ o, int n) {
    int id = blockIdx.x * blockDim.x + threadIdx.x;
    if (id < n) o[id] = a[id] - b[id];
}

__global__ void k_add(const float* __restrict__ a, const float* __restrict__ b,
                      float* __restrict__ o, int n) {
    int id = blockIdx.x * blockDim.x + threadIdx.x;
    if (id < n) o[id] = a[id] + b[id];
}

// (B, R, C) -> (B, C, R)
__global__ void k_transpose(const float* __restrict__ in, float* __restrict__ out,
                            int R, int Cc, int total) {
    int id = blockIdx.x * blockDim.x + threadIdx.x;
    if (id >= total) return;
    int c = id % Cc;
    int t = id / Cc;
    int r = t % R;
    int b = t / R;
    out[((size_t)b * Cc + c) * R + r] = in[id];
}

// concat [x1;x2;x3;x4;f1] along channels -> (B, 1280, 256)
__global__ void k_concat5(const float* __restrict__ x1, const float* __restrict__ x2,
                          const float* __restrict__ x3, const float* __restrict__ x4,
                          const float* __restrict__ f1, float* __restrict__ out, int total) {
    int id = blockIdx.x * blockDim.x + threadIdx.x;
    if (id >= total) return;
    int n = id % 256;
    int t = id / 256;
    int ch = t % 1280;
    int b = t / 1280;
    const float* src;
    int c;
    if (ch < 256) { src = x1; c = ch; }
    else if (ch < 512) { src = x2; c = ch - 256; }
    else if (ch < 768) { src = x3; c = ch - 512; }
    else if (ch < 1024) { src = x4; c = ch - 768; }
    else { src = f1; c = ch - 1024; }
    out[id] = src[((size_t)b * 256 + c) * 256 + n];
}

__global__ void k_maxpool(const float* __restrict__ in, float* __restrict__ out,
                          int Cc, int Np, int total) {
    int id = blockIdx.x * blockDim.x + threadIdx.x;
    if (id >= total) return;
    const float* p = in + (size_t)id * Np;
    float mx = -3.4e38f;
    for (int s = 0; s < Np; ++s) mx = fmaxf(mx, p[s]);
    out[id] = mx;
}

// ---------------------------------------------------------------------------
// Host-side orchestration
// ---------------------------------------------------------------------------
struct BNp { const float *g, *b, *m, *v; };
struct SAp { const float *qk, *vw, *vb, *tw, *tb; BNp bn; const float *pw1, *pb1, *pw2, *pb2; };

static inline void gemm(hipStream_t st, const float* A, const float* Bm, float* D,
                        int M, int N, int K, int lda, int ldb, int ldd,
                        long sA, long sB, long sD, int batch, int tA, int tB,
                        const float* bias, const BNp* bn, int act) {
    dim3 grid((N + 15) / 16, (M + 31) / 32, batch);
    const float* g = bn ? bn->g : nullptr;
    const float* b = bn ? bn->b : nullptr;
    const float* m = bn ? bn->m : nullptr;
    const float* v = bn ? bn->v : nullptr;
    if (tA) {
        k_gemm<1, 0><<<grid, 32, 0, st>>>(A, Bm, D, M, N, K, lda, ldb, ldd, sA, sB, sD,
                                          bias, g, b, m, v, act);
    } else if (tB) {
        k_gemm<0, 1><<<grid, 32, 0, st>>>(A, Bm, D, M, N, K, lda, ldb, ldd, sA, sB, sD,
                                          bias, g, b, m, v, act);
    } else {
        k_gemm<0, 0><<<grid, 32, 0, st>>>(A, Bm, D, M, N, K, lda, ldb, ldd, sA, sB, sD,
                                          bias, g, b, m, v, act);
    }
}

extern "C" void kernel_launch(void* const* d_in, const int* in_sizes, int n_in,
                              void* d_out, int out_size, void* d_ws, size_t ws_size,
                              hipStream_t stream) {
    (void)in_sizes; (void)n_in; (void)out_size; (void)ws_size;
    int ii = 0;
    auto F = [&]() { return (const float*)d_in[ii++]; };
    auto BNP = [&]() { BNp r; r.g = F(); r.b = F(); r.m = F(); r.v = F(); return r; };
    auto SAP = [&]() {
        SAp s;
        s.qk = F(); s.vw = F(); s.vb = F(); s.tw = F(); s.tb = F();
        s.bn = BNP();
        s.pw1 = F(); s.pb1 = F(); s.pw2 = F(); s.pb2 = F();
        return s;
    };

    const float* x = F();
    const float* w1 = F();  BNp bn1 = BNP();
    const float* w2 = F();  BNp bn2 = BNP();
    const float* g0w1 = F(); BNp g0bn1 = BNP();
    const float* g0w2 = F(); BNp g0bn2 = BNP();
    const float* g1w1 = F(); BNp g1bn1 = BNP();
    const float* g1w2 = F(); BNp g1bn2 = BNP();
    const float* saw1 = F(); BNp sabn1 = BNP();
    const float* saw2 = F(); BNp sabn2 = BNP();
    SAp l1 = SAP(), l2 = SAP(), l3 = SAP(), l4 = SAP();
    const float* fusew = F(); BNp fusebn = BNP();
    const float* l1w = F();   BNp bn6 = BNP();
    const float* l2w = F();   const float* l2b = F(); BNp bn7 = BNP();
    const float* l3w = F();   const float* l3b = F();

    // ---- workspace bump allocator ----
    char* wsp = (char*)d_ws;
    size_t off = 0;
    auto alloc = [&](size_t bytes) -> void* {
        void* p = wsp + off;
        off = (off + bytes + 255) & ~(size_t)255;
        return p;
    };
    const size_t FM = (size_t)BB * 256 * 256;  // sa-stage feature map elements
    float* feat0 = (float*)alloc((size_t)BB * NPTS * 64 * 4);
    int* fps1    = (int*)alloc((size_t)BB * M1 * 4);
    float* nxyz1 = (float*)alloc((size_t)BB * M1 * 3 * 4);
    int* knn1    = (int*)alloc((size_t)BB * M1 * 32 * 4);
    float* f0    = (float*)alloc((size_t)BB * M1 * 128 * 4);
    int* fps2    = (int*)alloc((size_t)BB * M2 * 4);
    float* nxyz2 = (float*)alloc((size_t)BB * M2 * 3 * 4);
    int* knn2    = (int*)alloc((size_t)BB * M2 * 32 * 4);
    float* f1pm  = (float*)alloc(FM * 4);
    float* f1cm  = (float*)alloc(FM * 4);
    float* hA    = (float*)alloc(FM * 4);
    float* hB    = (float*)alloc(FM * 4);
    float* x1b   = (float*)alloc(FM * 4);
    float* x2b   = (float*)alloc(FM * 4);
    float* x3b   = (float*)alloc(FM * 4);
    float* x4b   = (float*)alloc(FM * 4);
    float* xk    = (float*)alloc((size_t)BB * 64 * 256 * 4);
    float* xv    = (float*)alloc(FM * 4);
    float* energy= (float*)alloc(FM * 4);
    float* pos   = (float*)alloc(FM * 4);
    float* tmp   = (float*)alloc(FM * 4);
    float* xr    = (float*)alloc(FM * 4);
    float* catb  = (float*)alloc((size_t)BB * 1280 * 256 * 4);
    float* fout  = (float*)alloc((size_t)BB * 1024 * 256 * 4);
    float* pooled= (float*)alloc((size_t)BB * 1024 * 4);
    float* fc1   = (float*)alloc((size_t)BB * 512 * 4);
    float* fc2   = (float*)alloc((size_t)BB * 256 * 4);

    // ---- 1. input MLP ----
    {
        int total = BB * NPTS;
        k_inmlp<<<(total + 255) / 256, 256, 0, stream>>>(
            x, w1, bn1.g, bn1.b, bn1.m, bn1.v, w2, bn2.g, bn2.b, bn2.m, bn2.v,
            feat0, total);
    }

    // ---- 2. sample_and_group level 1 + local_op g0 ----
    k_fps<<<BB, 256, 0, stream>>>(x, NPTS, M1, fps1);
    k_gather3<<<(BB * M1 + 255) / 256, 256, 0, stream>>>(x, fps1, M1, NPTS, nxyz1, BB * M1);
    k_knn<<<(BB * M1 + 127) / 128, 128, 0, stream>>>(nxyz1, x, M1, NPTS, knn1, BB * M1);
    k_localop<<<BB * M1, 128, 0, stream>>>(
        feat0, NPTS, 64, M1, fps1, knn1,
        g0w1, g0bn1.g, g0bn1.b, g0bn1.m, g0bn1.v,
        g0w2, g0bn2.g, g0bn2.b, g0bn2.m, g0bn2.v, f0);

    // ---- 3. sample_and_group level 2 + local_op g1 ----
    k_fps<<<BB, 256, 0, stream>>>(nxyz1, M1, M2, fps2);
    k_gather3<<<(BB * M2 + 255) / 256, 256, 0, stream>>>(nxyz1, fps2, M2, M1, nxyz2, BB * M2);
    k_knn<<<(BB * M2 + 127) / 128, 128, 0, stream>>>(nxyz2, nxyz1, M2, M1, knn2, BB * M2);
    k_localop<<<BB * M2, 128, 0, stream>>>(
        f0, M1, 128, M2, fps2, knn2,
        g1w1, g1bn1.g, g1bn1.b, g1bn1.m, g1bn1.v,
        g1w2, g1bn2.g, g1bn2.b, g1bn2.m, g1bn2.v, f1pm);

    // ---- 4. f1 point-major -> channel-major ----
    k_transpose<<<((int)FM + 255) / 256, 256, 0, stream>>>(f1pm, f1cm, 256, 256, (int)FM);

    // ---- 5. SA pre-convs ----
    gemm(stream, saw1, f1cm, hA, 256, 256, 256, 256, 256, 256, 0, 65536, 65536, BB,
         0, 0, nullptr, &sabn1, 1);
    gemm(stream, saw2, hA, hB, 256, 256, 256, 256, 256, 256, 0, 65536, 65536, BB,
         0, 0, nullptr, &sabn2, 1);

    // ---- 6. four SA layers ----
    auto run_sa = [&](const SAp& L, const float* xin, float* xout) {
        const int nFM = (int)FM;
        // xk = qk_w @ x
        gemm(stream, L.qk, xin, xk, 64, 256, 256, 256, 256, 256, 0, 65536, 64 * 256, BB,
             0, 0, nullptr, nullptr, 0);
        // xv = v_w @ x + v_b
        gemm(stream, L.vw, xin, xv, 256, 256, 256, 256, 256, 256, 0, 65536, 65536, BB,
             0, 0, L.vb, nullptr, 0);
        // energy = xk^T @ xk
        gemm(stream, xk, xk, energy, 256, 256, 64, 256, 256, 256,
             64 * 256, 64 * 256, 65536, BB, 1, 0, nullptr, nullptr, 0);
        // positional bias
        k_pos<<<(nFM + 255) / 256, 256, 0, stream>>>(nxyz2, L.pw1, L.pb1, L.pw2, L.pb2,
                                                     pos, 256, nFM);
        // att = softmax(energy + pos) then column-renormalize (in place)
        k_softmax<<<BB * 256, 256, 0, stream>>>(energy, pos, 256);
        k_colnorm<<<BB * 256, 256, 0, stream>>>(energy, 256);
        // xr = xv @ att
        gemm(stream, xv, energy, xr, 256, 256, 256, 256, 256, 256,
             65536, 65536, 65536, BB, 0, 0, nullptr, nullptr, 0);
        // tmp = x - xr
        k_sub<<<(nFM + 255) / 256, 256, 0, stream>>>(xin, xr, tmp, nFM);
        // xr = relu(bn(tw @ tmp + tb))
        gemm(stream, L.tw, tmp, xr, 256, 256, 256, 256, 256, 256,
             0, 65536, 65536, BB, 0, 0, L.tb, &L.bn, 1);
        // xout = x + xr
        k_add<<<(nFM + 255) / 256, 256, 0, stream>>>(xin, xr, xout, nFM);
    };
    run_sa(l1, hB, x1b);
    run_sa(l2, x1b, x2b);
    run_sa(l3, x2b, x3b);
    run_sa(l4, x3b, x4b);

    // ---- 7. concat + fuse conv + leaky + maxpool ----
    {
        int total = BB * 1280 * 256;
        k_concat5<<<(total + 255) / 256, 256, 0, stream>>>(x1b, x2b, x3b, x4b, f1cm,
                                                           catb, total);
    }
    gemm(stream, fusew, catb, fout, 1024, 256, 1280, 1280, 256, 256,
         0, (long)1280 * 256, (long)1024 * 256, BB, 0, 0, nullptr, &fusebn, 2);
    {
        int total = BB * 1024;
        k_maxpool<<<(total + 255) / 256, 256, 0, stream>>>(fout, pooled, 1024, 256, total);
    }

    // ---- 8. classifier head ----
    gemm(stream, pooled, l1w, fc1, 32, 512, 1024, 1024, 1024, 512, 0, 0, 0, 1,
         0, 1, nullptr, &bn6, 1);
    gemm(stream, fc1, l2w, fc2, 32, 256, 512, 512, 512, 256, 0, 0, 0, 1,
         0, 1, l2b, &bn7, 1);
    gemm(stream, fc2, l3w, (float*)d_out, 32, 40, 256, 256, 256, 40, 0, 0, 0, 1,
         0, 1, l3b, nullptr, 0);
}